// Message_Module_48644799595011
// MI455X (gfx1250) — compile-verified
//
#include <hip/hip_runtime.h>
#include <stdint.h>

#ifndef __has_builtin
#define __has_builtin(x) 0
#endif

#define D_FEAT 64
#define TILE   256          // edges staged per block-iteration
#define BLOCK  256          // 8 wave32 per block
#define EPW    (TILE / (BLOCK / 32))   // 32 edges processed per wave per tile
#define MAX_GRID 640        // persistent blocks; >=2 tiles each for pipelining

// ---------------------------------------------------------------------------
// PROBE (never launched, FIRST in file so the disasm snippet reveals the
// lowering of the f32 atomic add and the prefetch used by the hot loop)
// ---------------------------------------------------------------------------
__global__ __launch_bounds__(32) void atomic_lowering_probe(
    float* __restrict__ o, const float* __restrict__ q) {
  __builtin_prefetch(q + threadIdx.x + 64, 0, 3);
  float v = q[threadIdx.x];
  __hip_atomic_fetch_add(o + threadIdx.x, v, __ATOMIC_RELAXED,
                         __HIP_MEMORY_SCOPE_AGENT);
}

// ---------------------------------------------------------------------------
// gfx1250 async global->LDS staging (ASYNCcnt path), with fallbacks
// ---------------------------------------------------------------------------
#if __has_builtin(__builtin_amdgcn_global_load_async_to_lds_b32)
  #define HAVE_ASYNC_BUILTIN 1
#else
  #define HAVE_ASYNC_BUILTIN 0
#endif

#if HAVE_ASYNC_BUILTIN
  #define AS1 __attribute__((address_space(1)))
  #define AS3 __attribute__((address_space(3)))
  typedef AS1 int g_i32_t;
  typedef AS3 int l_i32_t;
  __device__ __forceinline__ void async_b32(const void* g, void* l) {
    __builtin_amdgcn_global_load_async_to_lds_b32(
        (g_i32_t*)(uintptr_t)g, (l_i32_t*)(uintptr_t)l, 0, 0);
  }
#else
  // inline-asm form per CDNA5 ISA
  __device__ __forceinline__ void async_b32(const void* g, void* l) {
    unsigned laddr = (unsigned)(uintptr_t)l;   // flat LDS ptr: addr[31:0] is LDS offset
    asm volatile("global_load_async_to_lds_b32 %0, %1, off"
                 :: "v"(laddr), "v"((uint64_t)(uintptr_t)g)
                 : "memory");
  }
#endif

__device__ __forceinline__ void wait_async_zero() {
#if __has_builtin(__builtin_amdgcn_s_wait_asynccnt)
  __builtin_amdgcn_s_wait_asynccnt(0);
#else
  asm volatile("s_wait_asynccnt 0" ::: "memory");
#endif
}

// ---------------------------------------------------------------------------
// stage one tile of edge tuples into LDS (one edge per thread, 3 x b32)
// ---------------------------------------------------------------------------
__device__ __forceinline__ void stage_tile(const int* __restrict__ src,
                                           const int* __restrict__ dst,
                                           const float* __restrict__ w,
                                           int base, int n_edges,
                                           int* bs, int* bd, float* bw, int tid) {
  int e = base + tid;
  if (e < n_edges) {             // EXEC-masked; async VMEM honors per-lane EXEC
    async_b32(src + e, bs + tid);
    async_b32(dst + e, bd + tid);
    async_b32(w   + e, bw + tid);
  }
}

// ---------------------------------------------------------------------------
// process one edge: wave32, float2 per lane => full 256B row per edge
// ---------------------------------------------------------------------------
__device__ __forceinline__ void do_edge(const float* __restrict__ queue,
                                        float* __restrict__ out,
                                        int s, int d, float w, int s_next,
                                        int lane, bool prefetch) {
  if (prefetch) {
    __builtin_prefetch(queue + (size_t)s_next * D_FEAT + 2 * lane, 0, 3);
  }
  const float2 q = *reinterpret_cast<const float2*>(
      queue + (size_t)s * D_FEAT + 2 * lane);
  float* o = out + (size_t)d * D_FEAT + 2 * lane;
  // non-returning relaxed agent-scope f32 adds -> global_atomic_add_f32
  __hip_atomic_fetch_add(o,     q.x * w, __ATOMIC_RELAXED, __HIP_MEMORY_SCOPE_AGENT);
  __hip_atomic_fetch_add(o + 1, q.y * w, __ATOMIC_RELAXED, __HIP_MEMORY_SCOPE_AGENT);
}

// ---------------------------------------------------------------------------
// main scatter kernel: persistent blocks, double-buffered async edge staging
// ---------------------------------------------------------------------------
__global__ __launch_bounds__(BLOCK) void gnn_scatter_kernel(
    const float* __restrict__ queue,
    const float* __restrict__ weight,
    const int*   __restrict__ src,
    const int*   __restrict__ dst,
    float*       __restrict__ out,
    int n_edges) {
  __shared__ int   s_src[2][TILE];
  __shared__ int   s_dst[2][TILE];
  __shared__ float s_w[2][TILE];

  const int tid  = threadIdx.x;
  const int lane = tid & 31;
  const int wv   = tid >> 5;
  const int numTiles = (n_edges + TILE - 1) / TILE;
  int tile = blockIdx.x;
  if (tile >= numTiles) return;

  int cur = 0;
  stage_tile(src, dst, weight, tile * TILE, n_edges,
             s_src[0], s_dst[0], s_w[0], tid);
  wait_async_zero();
  __syncthreads();

  for (; tile < numTiles; tile += gridDim.x) {
    const int nextTile = tile + gridDim.x;
    if (nextTile < numTiles) {
      // overlap: fetch tile t+stride while computing tile t
      stage_tile(src, dst, weight, nextTile * TILE, n_edges,
                 s_src[cur ^ 1], s_dst[cur ^ 1], s_w[cur ^ 1], tid);
    }

    const int  base = tile * TILE;
    const int* bs = s_src[cur];
    const int* bd = s_dst[cur];
    const float* bw = s_w[cur];

    if (base + TILE <= n_edges) {
      // fast path: full tile, no per-edge bounds checks
      #pragma unroll 4
      for (int i = 0; i < EPW; ++i) {
        const int t = wv * EPW + i;
        const bool pf = (i + 1 < EPW);
        do_edge(queue, out, bs[t], bd[t], bw[t],
                pf ? bs[t + 1] : 0, lane, pf);
      }
    } else {
      // tail tile
      for (int i = 0; i < EPW; ++i) {
        const int t = wv * EPW + i;
        if (base + t >= n_edges) break;
        do_edge(queue, out, bs[t], bd[t], bw[t], 0, lane, false);
      }
    }

    wait_async_zero();   // my async writes to the other buffer are done
    __syncthreads();     // ...and visible to all waves; all reads of cur done
    cur ^= 1;
  }
}

// ---------------------------------------------------------------------------
// zero-fill output (harness poisons d_out with 0xAA before timing)
// ---------------------------------------------------------------------------
__global__ void zero_f4_kernel(float4* __restrict__ out4, int n4) {
  int i = blockIdx.x * blockDim.x + threadIdx.x;
  if (i < n4) {
    float4 z = {0.f, 0.f, 0.f, 0.f};
    out4[i] = z;
  }
}

__global__ void zero_tail_kernel(float* __restrict__ out, int start, int n) {
  int i = start + (int)threadIdx.x;
  if (i < n) out[i] = 0.f;
}

// ---------------------------------------------------------------------------
extern "C" void kernel_launch(void* const* d_in, const int* in_sizes, int n_in,
                              void* d_out, int out_size, void* d_ws, size_t ws_size,
                              hipStream_t stream) {
  const float* queue  = (const float*)d_in[0];
  const float* weight = (const float*)d_in[1];
  const int*   src    = (const int*)d_in[2];
  const int*   dst    = (const int*)d_in[3];
  float*       out    = (float*)d_out;
  const int n_edges = in_sizes[2];

  // 1) zero the output accumulator
  {
    int n4 = out_size / 4;
    if (n4 > 0) {
      int grid = (n4 + 255) / 256;
      hipLaunchKernelGGL(zero_f4_kernel, dim3(grid), dim3(256), 0, stream,
                         (float4*)out, n4);
    }
    int rem = out_size - n4 * 4;
    if (rem > 0) {
      hipLaunchKernelGGL(zero_tail_kernel, dim3(1), dim3(64), 0, stream,
                         out, n4 * 4, out_size);
    }
  }

  // 2) gather * weight, scatter-add via hardware f32 atomics
  {
    int numTiles = (n_edges + TILE - 1) / TILE;
    int grid = numTiles < MAX_GRID ? numTiles : MAX_GRID;
    if (grid < 1) grid = 1;
    hipLaunchKernelGGL(gnn_scatter_kernel, dim3(grid), dim3(BLOCK), 0, stream,
                       queue, weight, src, dst, out, n_edges);
  }
}